// Encoder_MLP_76390288326988
// MI455X (gfx1250) — compile-verified
//
#include <hip/hip_runtime.h>
#include <hip/hip_bf16.h>

// Problem constants (match reference)
#define NA   4095          // number of actions (pad_id == NA)
#define NV   (NA + 1)      // vocab = 4096
#define NT   256           // max gates
#define NB   64            // batch
#define NHID 256
#define NOUT 256

typedef __attribute__((ext_vector_type(2))) float v2f;
typedef __attribute__((ext_vector_type(8))) float v8f;

// ---------------------------------------------------------------------------
// Stage 1a: gather-and-sum partials.
// grid = (8 chunks, 64 batch), block = 256 threads (8 waves).
// Block (ch, b) sums 32 rows of W1 selected by gate_seq[b, ch*32 .. ch*32+31].
// Thread tid owns hidden unit tid -> each row read is a fully coalesced 1KB
// burst across the block. Unroll 8 => 8 independent loads in flight per lane.
// ---------------------------------------------------------------------------
__global__ void gather_partial(const int* __restrict__ gate,
                               const float* __restrict__ W1,
                               float* __restrict__ partial) {
    const int ch  = blockIdx.x;        // 0..7
    const int b   = blockIdx.y;        // 0..63
    const int tid = threadIdx.x;       // 0..255
    const int t0  = ch * 32;

    float acc = 0.0f;
#pragma unroll 8
    for (int i = 0; i < 32; ++i) {
        const int t = t0 + i;
        int g = gate[b * NT + t];      // uniform across block -> scalar load
        if (g < 0) g = NA;             // pad mapping, as in reference
        const size_t row = (size_t)t * NV + (size_t)g;
        acc += W1[row * NHID + tid];
    }
    partial[((size_t)b * 8 + ch) * NHID + tid] = acc;
}

// ---------------------------------------------------------------------------
// Stage 1b: fold the 8 partials, add b1, ReLU.
// grid = 64, block = 256.
// ---------------------------------------------------------------------------
__global__ void reduce_relu(const float* __restrict__ partial,
                            const float* __restrict__ b1,
                            float* __restrict__ h1) {
    const int b   = blockIdx.x;
    const int tid = threadIdx.x;
    float acc = b1[tid];
#pragma unroll
    for (int ch = 0; ch < 8; ++ch)
        acc += partial[((size_t)b * 8 + ch) * NHID + tid];
    h1[(size_t)b * NHID + tid] = fmaxf(acc, 0.0f);
}

// ---------------------------------------------------------------------------
// Stages 2 & 3: C[64x256] = act(A[64x256] @ B[256x256] + bias)
// using V_WMMA_F32_16X16X4_F32 (full fp32 matrix path, matches reference
// precision). One wave per 16x16 C tile: 4 M-tiles x 16 N-tiles = 64 waves
// = 8 blocks x 8 waves. K advances 4 per WMMA (64 wmma issues per wave).
//
// f32 fragment layout (ISA 7.12.2):
//   A 16x4 : lane m=lane%16; VGPR0/1 = K {kh, kh+1}, kh = (lane<16)?0:2
//   B 4x16 : lane n=lane%16; VGPR0/1 = K {kh, kh+1}  (rows striped)
//   C/D    : VGPR r = row (mt*16 + r + (lane<16?0:8)), col = nt*16 + lane%16
// ---------------------------------------------------------------------------
__global__ void gemm64x256_wmma(const float* __restrict__ Amat,
                                const float* __restrict__ Bmat,
                                const float* __restrict__ bias,
                                float* __restrict__ Cmat,
                                const int do_relu) {
    const int K = NHID, N = NOUT;

    const int wave = (int)((blockIdx.x * blockDim.x + threadIdx.x) >> 5); // 0..63
    const int lane = (int)(threadIdx.x & 31);
    const int mt = wave >> 4;          // 0..3
    const int nt = wave & 15;          // 0..15

    const int l16   = lane & 15;
    const int khalf = (lane < 16) ? 0 : 2;
    const int mrow  = mt * 16 + l16;   // A row this lane reads
    const int ncol  = nt * 16 + l16;   // B/C column this lane owns

    const float* Aptr = Amat + (size_t)mrow * K + khalf;   // 8B aligned
    const float* Bptr = Bmat + (size_t)khalf * N + ncol;

    v8f acc = {};
#pragma unroll 4
    for (int k0 = 0; k0 < K; k0 += 4) {
        v2f a = *(const v2f*)(Aptr + k0);                  // global_load_b64
        v2f b;
        b.x = Bptr[(size_t)k0 * N];
        b.y = Bptr[(size_t)(k0 + 1) * N];
        acc = __builtin_amdgcn_wmma_f32_16x16x4_f32(
            /*neg_a=*/false, a, /*neg_b=*/false, b,
            /*c_mod=*/(short)0, acc, /*reuse_a=*/false, /*reuse_b=*/false);
    }

    const int mbase = mt * 16 + ((lane < 16) ? 0 : 8);
    const float bn = bias[ncol];
#pragma unroll
    for (int r = 0; r < 8; ++r) {
        float v = acc[r] + bn;
        if (do_relu) v = fmaxf(v, 0.0f);
        Cmat[(size_t)(mbase + r) * N + ncol] = v;
    }
}

// ---------------------------------------------------------------------------
// Launcher. Workspace layout (floats):
//   [0, 131072)          partial sums  (64 batches x 8 chunks x 256)
//   [131072, 147456)     h1
//   [147456, 163840)     h2
// Total 640 KB.
// ---------------------------------------------------------------------------
extern "C" void kernel_launch(void* const* d_in, const int* in_sizes, int n_in,
                              void* d_out, int out_size, void* d_ws, size_t ws_size,
                              hipStream_t stream) {
    const int*   gate = (const int*)d_in[0];
    const float* W1   = (const float*)d_in[1];
    const float* b1   = (const float*)d_in[2];
    const float* W2   = (const float*)d_in[3];
    const float* b2   = (const float*)d_in[4];
    const float* W3   = (const float*)d_in[5];
    const float* b3   = (const float*)d_in[6];
    float*       out  = (float*)d_out;

    float* ws      = (float*)d_ws;
    float* partial = ws;                        // 64*8*256
    float* h1      = ws + (size_t)NB * 8 * NHID;
    float* h2      = h1 + (size_t)NB * NHID;

    gather_partial<<<dim3(8, NB), 256, 0, stream>>>(gate, W1, partial);
    reduce_relu   <<<NB, 256, 0, stream>>>(partial, b1, h1);
    gemm64x256_wmma<<<8, 256, 0, stream>>>(h1, W2, b2, h2, 1);
    gemm64x256_wmma<<<8, 256, 0, stream>>>(h2, W3, b3, out, 0);
}